// GAT_87540023427950
// MI455X (gfx1250) — compile-verified
//
#include <hip/hip_runtime.h>
#include <hip/hip_bf16.h>

typedef float v2f __attribute__((ext_vector_type(2)));
typedef float v8f __attribute__((ext_vector_type(8)));

#define NEG_SLOPE 0.2f

// ---------------------------------------------------------------------------
// Collapse projection weight against attention vector:
//   out[f,h] = sum_d W[f, h*D + d] * a[h, d]      (out is [F x H], row-major)
// ---------------------------------------------------------------------------
__global__ void collapse_w_kernel(const float* __restrict__ W,
                                  const float* __restrict__ a,
                                  float* __restrict__ out,
                                  int F, int H, int D) {
  int idx = blockIdx.x * blockDim.x + threadIdx.x;
  if (idx >= F * H) return;
  int f = idx / H, h = idx % H;
  float s = 0.f;
  for (int d = 0; d < D; ++d) s += W[f * (H * D) + h * D + d] * a[h * D + d];
  out[f * H + h] = s;
}

// Zero-pad weights [F x Cin] -> [F x Cpad]
__global__ void pad_w_kernel(const float* __restrict__ W, float* __restrict__ out,
                             int F, int Cin, int Cpad) {
  int idx = blockIdx.x * blockDim.x + threadIdx.x;
  if (idx >= F * Cpad) return;
  int f = idx / Cpad, c = idx % Cpad;
  out[idx] = (c < Cin) ? W[f * Cin + c] : 0.f;
}

// ---------------------------------------------------------------------------
// WMMA f32 GEMM: C[M x NT*16] = A[M x 128] * B[128 x NT*16]  (all row-major)
// One wave computes a 16-row stripe (all NT column tiles), reusing the A
// fragment across tiles. Uses V_WMMA_F32_16X16X4_F32 (full fp32 precision).
// ---------------------------------------------------------------------------
template <int NT>
__global__ void wmma_gemm_f32_kernel(const float* __restrict__ A,
                                     const float* __restrict__ B,
                                     float* __restrict__ C, int M) {
  const int K = 128;
  const int ldb = NT * 16, ldc = NT * 16;
  int gid  = blockIdx.x * blockDim.x + threadIdx.x;
  int wave = gid >> 5;
  int lane = gid & 31;
  int m0 = wave * 16;
  if (m0 >= M) return;                 // wave-uniform: EXEC stays all-ones
  int half = lane >> 4;                // selects K pair (A) / K pair (B)
  int r    = lane & 15;                // M index (A) / N index (B,D)

  v8f acc[NT];
  const v8f vzero = {0.f, 0.f, 0.f, 0.f, 0.f, 0.f, 0.f, 0.f};
#pragma unroll
  for (int t = 0; t < NT; ++t) acc[t] = vzero;

  const float* Arow = A + (size_t)(m0 + r) * K;
  for (int k0 = 0; k0 < K; k0 += 4) {
    int ka = k0 + 2 * half;
    v2f a;
    a.x = Arow[ka];
    a.y = Arow[ka + 1];
#pragma unroll
    for (int t = 0; t < NT; ++t) {
      v2f b;
      b.x = B[(size_t)ka * ldb + t * 16 + r];
      b.y = B[(size_t)(ka + 1) * ldb + t * 16 + r];
      acc[t] = __builtin_amdgcn_wmma_f32_16x16x4_f32(
          /*neg_a=*/false, a, /*neg_b=*/false, b,
          /*c_mod=*/(short)0, acc[t], /*reuse_a=*/false, /*reuse_b=*/false);
    }
  }
#pragma unroll
  for (int t = 0; t < NT; ++t) {
#pragma unroll
    for (int j = 0; j < 8; ++j) {
      int row = m0 + j + 8 * half;
      C[(size_t)row * ldc + t * 16 + r] = acc[t][j];
    }
  }
}

// ---------------------------------------------------------------------------
// el[n,h] = x[n,:] @ wl[:,h],  er[n,h] = x[n,:] @ wr[:,h]   (one wave / node)
// ---------------------------------------------------------------------------
__global__ void node_elr_kernel(const float* __restrict__ X,
                                const float* __restrict__ wl,
                                const float* __restrict__ wr,
                                float* __restrict__ el, float* __restrict__ er,
                                int N) {
  const int F = 128, H = 4;
  int gid = blockIdx.x * blockDim.x + threadIdx.x;
  int node = gid >> 5, lane = gid & 31;
  if (node >= N) return;
  float aL[H] = {0.f, 0.f, 0.f, 0.f};
  float aR[H] = {0.f, 0.f, 0.f, 0.f};
  const float* xp = X + (size_t)node * F;
  for (int f = lane; f < F; f += 32) {
    float x = xp[f];
#pragma unroll
    for (int h = 0; h < H; ++h) {
      aL[h] += x * wl[f * H + h];
      aR[h] += x * wr[f * H + h];
    }
  }
#pragma unroll
  for (int h = 0; h < H; ++h)
    for (int off = 16; off > 0; off >>= 1) {
      aL[h] += __shfl_xor(aL[h], off, 32);
      aR[h] += __shfl_xor(aR[h], off, 32);
    }
  if (lane == 0) {
#pragma unroll
    for (int h = 0; h < H; ++h) {
      el[node * H + h] = aL[h];
      er[node * H + h] = aR[h];
    }
  }
}

// Monotone float<->u32 key for atomic float max via atomicMax(u32)
__device__ __forceinline__ unsigned fkey(float f) {
  unsigned u = __float_as_uint(f);
  return (u & 0x80000000u) ? ~u : (u | 0x80000000u);
}
__device__ __forceinline__ float funkey(unsigned k) {
  unsigned u = (k & 0x80000000u) ? (k ^ 0x80000000u) : ~k;
  return __uint_as_float(u);
}

// Phase 1: e = leaky_relu(el[src]+er[dst]); segment-max into nmaxKey[dst]
__global__ void edge_attn_max_kernel(const int* __restrict__ src,
                                     const int* __restrict__ dst,
                                     const float* __restrict__ el,
                                     const float* __restrict__ er,
                                     float* __restrict__ eBuf,
                                     unsigned* __restrict__ nmaxKey, int E) {
  const int H = 4;
  int e = blockIdx.x * blockDim.x + threadIdx.x;
  if (e >= E) return;
  int s = src[e], d = dst[e];
#pragma unroll
  for (int h = 0; h < H; ++h) {
    float v = el[s * H + h] + er[d * H + h];
    v = (v > 0.f) ? v : NEG_SLOPE * v;
    eBuf[(size_t)e * H + h] = v;
    atomicMax(&nmaxKey[d * H + h], fkey(v));
  }
}

// Phase 2: ee = exp(e - emax[dst]); segment-sum into denom[dst]
__global__ void edge_exp_sum_kernel(const int* __restrict__ dst,
                                    const unsigned* __restrict__ nmaxKey,
                                    float* __restrict__ eBuf,
                                    float* __restrict__ denom, int E) {
  const int H = 4;
  int e = blockIdx.x * blockDim.x + threadIdx.x;
  if (e >= E) return;
  int d = dst[e];
#pragma unroll
  for (int h = 0; h < H; ++h) {
    float m = funkey(nmaxKey[d * H + h]);
    float ee = __expf(eBuf[(size_t)e * H + h] - m);
    eBuf[(size_t)e * H + h] = ee;
    atomicAdd(&denom[d * H + h], ee);
  }
}

// Phase 3 (layer 1): rst[dst] += alpha * fs[src], row = 128 floats, wave/edge
__global__ void edge_aggregate128_kernel(const int* __restrict__ src,
                                         const int* __restrict__ dst,
                                         const float* __restrict__ eBuf,
                                         const float* __restrict__ denom,
                                         const float* __restrict__ fs,
                                         float* __restrict__ rst, int E) {
  const int H = 4;
  int gid = blockIdx.x * blockDim.x + threadIdx.x;
  int e = gid >> 5, lane = gid & 31;
  if (e >= E) return;
  int s = src[e], d = dst[e];
  float alpha[H];
#pragma unroll
  for (int h = 0; h < H; ++h)
    alpha[h] = eBuf[(size_t)e * H + h] / denom[d * H + h];
  const float* fsp = fs + (size_t)s * 128;
  float* rp = rst + (size_t)d * 128;
#pragma unroll
  for (int j = 0; j < 4; ++j) {       // elem = lane + 32*j, head = j
    int elem = lane + 32 * j;
    atomicAdd(&rp[elem], alpha[j] * fsp[elem]);
  }
}

// Phase 3 (layer 2): padded row LD, valid HC=H*C elems, head = elem / C
__global__ void edge_aggregate_pad_kernel(const int* __restrict__ src,
                                          const int* __restrict__ dst,
                                          const float* __restrict__ eBuf,
                                          const float* __restrict__ denom,
                                          const float* __restrict__ fs,
                                          float* __restrict__ rst, int E,
                                          int C, int HC, int LD) {
  const int H = 4;
  int gid = blockIdx.x * blockDim.x + threadIdx.x;
  int e = gid >> 5, lane = gid & 31;
  if (e >= E) return;
  int s = src[e], d = dst[e];
  float alpha[H];
#pragma unroll
  for (int h = 0; h < H; ++h)
    alpha[h] = eBuf[(size_t)e * H + h] / denom[d * H + h];
  const float* fsp = fs + (size_t)s * LD;
  float* rp = rst + (size_t)d * LD;
  for (int elem = lane; elem < HC; elem += 32) {
    int h = elem / C;
    atomicAdd(&rp[elem], alpha[h] * fsp[elem]);
  }
}

// Layer-1 epilogue: h = elu(rst + b)  (in place)
__global__ void bias_elu_kernel(float* __restrict__ rst,
                                const float* __restrict__ b, int total, int F) {
  int idx = blockIdx.x * blockDim.x + threadIdx.x;
  if (idx >= total) return;
  float v = rst[idx] + b[idx % F];
  rst[idx] = (v > 0.f) ? v : expm1f(v);
}

// Final: mean over heads of (rst2 + b2), then log_softmax over C. Wave/node.
__global__ void head_mean_logsoftmax_kernel(const float* __restrict__ rst2,
                                            const float* __restrict__ b2,
                                            float* __restrict__ out, int N,
                                            int C, int LD) {
  const int H = 4;
  int gid = blockIdx.x * blockDim.x + threadIdx.x;
  int node = gid >> 5, lane = gid & 31;
  if (node >= N) return;
  const float* rp = rst2 + (size_t)node * LD;
  int c0 = lane, c1 = lane + 32;
  bool v1ok = (c1 < C);
  float v0 = 0.f, v1 = 0.f;
#pragma unroll
  for (int h = 0; h < H; ++h) {
    v0 += rp[h * C + c0] + b2[h * C + c0];
    if (v1ok) v1 += rp[h * C + c1] + b2[h * C + c1];
  }
  v0 *= 0.25f;
  v1 *= 0.25f;
  float m = fmaxf(v0, v1ok ? v1 : -3.4e38f);
  for (int off = 16; off > 0; off >>= 1) m = fmaxf(m, __shfl_xor(m, off, 32));
  float se = __expf(v0 - m) + (v1ok ? __expf(v1 - m) : 0.f);
  for (int off = 16; off > 0; off >>= 1) se += __shfl_xor(se, off, 32);
  float lse = m + __logf(se);
  out[(size_t)node * C + c0] = v0 - lse;
  if (v1ok) out[(size_t)node * C + c1] = v1 - lse;
}

// ---------------------------------------------------------------------------
extern "C" void kernel_launch(void* const* d_in, const int* in_sizes, int n_in,
                              void* d_out, int out_size, void* d_ws,
                              size_t ws_size, hipStream_t stream) {
  (void)n_in; (void)out_size; (void)ws_size;
  const float* x   = (const float*)d_in[0];
  const int*   src = (const int*)d_in[1];
  const int*   dst = (const int*)d_in[2];
  const float* W1s = (const float*)d_in[3];
  const float* W1d = (const float*)d_in[4];
  const float* al1 = (const float*)d_in[5];
  const float* ar1 = (const float*)d_in[6];
  const float* b1  = (const float*)d_in[7];
  const float* W2s = (const float*)d_in[8];
  const float* W2d = (const float*)d_in[9];
  const float* al2 = (const float*)d_in[10];
  const float* ar2 = (const float*)d_in[11];
  const float* b2  = (const float*)d_in[12];

  const int F = 128, H = 4, D = 32;
  const int N = in_sizes[0] / F;        // 50000
  const int E = in_sizes[1];            // 800000
  const int C = in_sizes[12] / H;       // 47
  const int HC = H * C;                 // 188
  const int HCP = ((HC + 15) / 16) * 16; // 192

  // ---- workspace layout (floats) ----
  float* ws = (float*)d_ws;
  size_t o = 0;
  float* wl1  = ws + o; o += (size_t)F * H;
  float* wr1  = ws + o; o += (size_t)F * H;
  float* wl2  = ws + o; o += (size_t)F * H;
  float* wr2  = ws + o; o += (size_t)F * H;
  float* W2sp = ws + o; o += (size_t)F * HCP;
  float* fs1  = ws + o; o += (size_t)N * F;
  float* el1  = ws + o; o += (size_t)N * H;
  float* er1  = ws + o; o += (size_t)N * H;
  unsigned* nmax1 = (unsigned*)(ws + o); o += (size_t)N * H;
  float* den1 = ws + o; o += (size_t)N * H;
  float* rst1 = ws + o; o += (size_t)N * F;      // becomes h1 after epilogue
  float* fs2  = ws + o; o += (size_t)N * HCP;
  float* el2  = ws + o; o += (size_t)N * H;
  float* er2  = ws + o; o += (size_t)N * H;
  unsigned* nmax2 = (unsigned*)(ws + o); o += (size_t)N * H;
  float* den2 = ws + o; o += (size_t)N * H;
  float* rst2 = ws + o; o += (size_t)N * HCP;
  float* eBuf = ws + o; o += (size_t)E * H;      // shared by both layers

  // ---- tiny weight preprocessing ----
  collapse_w_kernel<<<(F * H + 255) / 256, 256, 0, stream>>>(W1s, al1, wl1, F, H, D);
  collapse_w_kernel<<<(F * H + 255) / 256, 256, 0, stream>>>(W1d, ar1, wr1, F, H, D);
  collapse_w_kernel<<<(F * H + 255) / 256, 256, 0, stream>>>(W2s, al2, wl2, F, H, C);
  collapse_w_kernel<<<(F * H + 255) / 256, 256, 0, stream>>>(W2d, ar2, wr2, F, H, C);
  pad_w_kernel<<<(F * HCP + 255) / 256, 256, 0, stream>>>(W2s, W2sp, F, HC, HCP);

  // ---- zero accumulators (graph-capturable memset nodes) ----
  hipMemsetAsync(nmax1, 0, (size_t)N * H * 4, stream);
  hipMemsetAsync(den1,  0, (size_t)N * H * 4, stream);
  hipMemsetAsync(rst1,  0, (size_t)N * F * 4, stream);
  hipMemsetAsync(nmax2, 0, (size_t)N * H * 4, stream);
  hipMemsetAsync(den2,  0, (size_t)N * H * 4, stream);
  hipMemsetAsync(rst2,  0, (size_t)N * HCP * 4, stream);

  const int gemmWaves = (N + 15) / 16;
  const int gemmBlocks = (gemmWaves * 32 + 127) / 128;
  const int edgeBlocks256 = (E + 255) / 256;
  const int edgeWaveBlocks = (E * 32 + 127) / 128;
  const int nodeWaveBlocks = (N * 32 + 127) / 128;

  // ================= layer 1 =================
  wmma_gemm_f32_kernel<8><<<gemmBlocks, 128, 0, stream>>>(x, W1s, fs1, N);
  node_elr_kernel<<<nodeWaveBlocks, 128, 0, stream>>>(x, wl1, wr1, el1, er1, N);
  edge_attn_max_kernel<<<edgeBlocks256, 256, 0, stream>>>(src, dst, el1, er1, eBuf, nmax1, E);
  edge_exp_sum_kernel<<<edgeBlocks256, 256, 0, stream>>>(dst, nmax1, eBuf, den1, E);
  edge_aggregate128_kernel<<<edgeWaveBlocks, 128, 0, stream>>>(src, dst, eBuf, den1, fs1, rst1, E);
  bias_elu_kernel<<<(N * F + 255) / 256, 256, 0, stream>>>(rst1, b1, N * F, F);

  // ================= layer 2 =================
  wmma_gemm_f32_kernel<12><<<gemmBlocks, 128, 0, stream>>>(rst1, W2sp, fs2, N);
  node_elr_kernel<<<nodeWaveBlocks, 128, 0, stream>>>(rst1, wl2, wr2, el2, er2, N);
  edge_attn_max_kernel<<<edgeBlocks256, 256, 0, stream>>>(src, dst, el2, er2, eBuf, nmax2, E);
  edge_exp_sum_kernel<<<edgeBlocks256, 256, 0, stream>>>(dst, nmax2, eBuf, den2, E);
  edge_aggregate_pad_kernel<<<edgeWaveBlocks, 128, 0, stream>>>(src, dst, eBuf, den2, fs2, rst2, E, C, HC, HCP);
  head_mean_logsoftmax_kernel<<<nodeWaveBlocks, 128, 0, stream>>>(rst2, b2, (float*)d_out, N, C, HCP);
}